// spatialJumble_24137716203718
// MI455X (gfx1250) — compile-verified
//
#include <hip/hip_runtime.h>
#include <stdint.h>

// Problem constants (match the JAX reference).
#define BATCH 32
#define HEIGHT 56
#define WIDTH 56
#define CHANS 256
#define HW (HEIGHT * WIDTH)          // 3136
#define ROWS_PER_BLOCK 4             // 256 threads / (CHANS/4 lanes per row)

// out[b, p, ch] = x[b, perm[ch, p], ch] * mask[ch] + x[b, p, ch] * (1 - mask[ch])
//
// Grid: (HW/4, BATCH). Block: 256 threads.
//   lane group of 64 threads covers one (b,p) row, one float4 of channels each.
// Consecutive lanes cover consecutive channels -> coalesced b128 load/store on
// the direct path. Gather path is per-lane b32, issued only where mask != 0.
__global__ __launch_bounds__(256) void spatialJumble_kernel(
    const float* __restrict__ x,
    const float* __restrict__ mask,
    const int*   __restrict__ perm,
    float*       __restrict__ out)
{
    __shared__ float smask[CHANS];   // 1 KB

    // ---- CDNA5 async global->LDS staging of the per-channel mask ----------
    // Threads 0..63 (waves 0-1 of the block) each copy 16B: 64*16B = 1 KB.
    // VDST gives the LDS destination address; low 32 bits of the generic LDS pointer
    // are the workgroup-relative LDS byte offset, which is what VDST wants.
    if (threadIdx.x < CHANS / 4) {
        uint32_t lds_off = (uint32_t)(uintptr_t)(&smask[threadIdx.x * 4]);
        const float* gsrc = mask + threadIdx.x * 4;
        asm volatile("global_load_async_to_lds_b128 %0, %1, off"
                     :
                     : "v"(lds_off), "v"(gsrc)
                     : "memory");
    }
    asm volatile("s_wait_asynccnt 0" ::: "memory");
    __syncthreads();
    // -----------------------------------------------------------------------

    // Index math: no divides, no guards. b and batchBase are wave-uniform.
    const int b  = blockIdx.y;                                   // scalar
    const int p  = blockIdx.x * ROWS_PER_BLOCK + (threadIdx.x >> 6);
    const int ch = (threadIdx.x & 63) * 4;

    const size_t batchBase = (size_t)b * (HW * CHANS);           // x[b,0,0]
    const size_t rowBase   = batchBase + (size_t)p * CHANS + ch; // x[b,p,ch]

    // Coalesced direct-path read (global_load_b128) and LDS mask read (b128).
    float4 xd = *reinterpret_cast<const float4*>(x + rowBase);
    float4 m  = *reinterpret_cast<const float4*>(&smask[ch]);

    // perm is [CHANS, HW] row-major: perm[ch*HW + p] = source spatial index.
    // Issue all four perm loads up front (L2-resident, reused 32x across
    // batches) so the masked x-gathers below can overlap.
    const int* permp = perm + (size_t)ch * HW + p;
    int q0 = permp[0 * HW];
    int q1 = permp[1 * HW];
    int q2 = permp[2 * HW];
    int q3 = permp[3 * HW];

    // Gather path: only lanes/components with mask != 0 issue the random read.
    float s0 = 0.0f, s1 = 0.0f, s2 = 0.0f, s3 = 0.0f;
    if (m.x != 0.0f) s0 = x[batchBase + (size_t)q0 * CHANS + (ch + 0)];
    if (m.y != 0.0f) s1 = x[batchBase + (size_t)q1 * CHANS + (ch + 1)];
    if (m.z != 0.0f) s2 = x[batchBase + (size_t)q2 * CHANS + (ch + 2)];
    if (m.w != 0.0f) s3 = x[batchBase + (size_t)q3 * CHANS + (ch + 3)];

    // Exact reference blend: shuf*mask + x*(1-mask). With mask in {0,1} and
    // s_i = 0 on the skipped path this reproduces the reference bit-exactly.
    float4 o;
    o.x = s0 * m.x + xd.x * (1.0f - m.x);
    o.y = s1 * m.y + xd.y * (1.0f - m.y);
    o.z = s2 * m.z + xd.z * (1.0f - m.z);
    o.w = s3 * m.w + xd.w * (1.0f - m.w);

    // Coalesced b128 store.
    *reinterpret_cast<float4*>(out + rowBase) = o;
}

extern "C" void kernel_launch(void* const* d_in, const int* in_sizes, int n_in,
                              void* d_out, int out_size, void* d_ws, size_t ws_size,
                              hipStream_t stream)
{
    // setup_inputs() dict order: x (f32 [B,H,W,C]), mask (f32 [C]), perm (i32 [C,HW])
    const float* x    = (const float*)d_in[0];
    const float* mask = (const float*)d_in[1];
    const int*   perm = (const int*)d_in[2];
    float*       out  = (float*)d_out;

    dim3 block(256);
    dim3 grid(HW / ROWS_PER_BLOCK, BATCH);   // (784, 32) -> exact coverage

    hipLaunchKernelGGL(spatialJumble_kernel, grid, block, 0, stream,
                       x, mask, perm, out);
}